// FAGCN_13048110645521
// MI455X (gfx1250) — compile-verified
//
#include <hip/hip_runtime.h>
#include <hip/hip_bf16.h>

#define N_NODES  100000
#define N_EDGES  1600000
#define F_INDIM  256
#define D_OUT    64
#define N_LAYERS 2
#define EPS_RAW  0.3f

typedef float v2f __attribute__((ext_vector_type(2)));
typedef float v8f __attribute__((ext_vector_type(8)));

// -------------------------------------------------------------------------
// Kernel 1: h = relu(x @ W1^T + b1) using V_WMMA_F32_16X16X4_F32 (wave32).
// One wave computes a 16x64 stripe of h (4 accumulators): per K-step one
// shared A fragment feeds 4 WMMAs -> 1.25 VMEM loads per WMMA and x is
// streamed exactly once (non-temporal, keeps L2 for W1/h/agg).
//   A fragment (16x4 f32, 2 VGPR): lane l -> m = l&15, k-half = l>>4,
//       holds x[(rowBase+m)*256 + k + 2*khalf + {0,1}]          (float2)
//   B fragment (4x16 f32, 2 VGPR): lane l -> n = l&15, k-half = l>>4,
//       holds W1[(colBase+n)*256 + k + 2*khalf + {0,1}]         (float2)
//   C/D (16x16 f32, 8 VGPR): VGPR j, lane l -> row m = j + 8*(l>>4),
//       col n = l&15  (ISA 7.12.2 layout).
// Tiling exact: 100000 = 6250*16 rows, 64 = 4*16 cols, 256 = 64*4 K.
// -------------------------------------------------------------------------
__global__ __launch_bounds__(256)
void fagcn_gemm_relu(const float* __restrict__ x,
                     const float* __restrict__ W1,
                     const float* __restrict__ b1,
                     float* __restrict__ h) {
  const int lane = threadIdx.x & 31;
  const int wave = (blockIdx.x * blockDim.x + threadIdx.x) >> 5;  // row tile
  if (wave >= N_NODES / 16) return;    // wave-uniform: EXEC stays all-1s

  const int half = lane >> 4;          // k-half within fragment
  const int sub  = lane & 15;          // m (for A) / n (for B)
  const int rowBase = wave * 16;

  const float* aPtr = x  + (size_t)(rowBase + sub) * F_INDIM + 2 * half;
  const float* bPtr = W1 + (size_t)sub * F_INDIM + 2 * half;

  v8f acc0 = {}, acc1 = {}, acc2 = {}, acc3 = {};
#pragma unroll 2
  for (int k = 0; k < F_INDIM; k += 4) {
    v2f a  = __builtin_nontemporal_load((const v2f*)(aPtr + k));  // x: 1-use stream
    v2f b0 = *(const v2f*)(bPtr + k);                    // W1 cols  0..15
    v2f b1v = *(const v2f*)(bPtr + 16 * F_INDIM + k);    // W1 cols 16..31
    v2f b2 = *(const v2f*)(bPtr + 32 * F_INDIM + k);     // W1 cols 32..47
    v2f b3 = *(const v2f*)(bPtr + 48 * F_INDIM + k);     // W1 cols 48..63
    // (neg_a, A, neg_b, B, c_mod, C, reuse_a, reuse_b)
    acc0 = __builtin_amdgcn_wmma_f32_16x16x4_f32(false, a, false, b0,
                                                 (short)0, acc0, false, false);
    acc1 = __builtin_amdgcn_wmma_f32_16x16x4_f32(false, a, false, b1v,
                                                 (short)0, acc1, false, false);
    acc2 = __builtin_amdgcn_wmma_f32_16x16x4_f32(false, a, false, b2,
                                                 (short)0, acc2, false, false);
    acc3 = __builtin_amdgcn_wmma_f32_16x16x4_f32(false, a, false, b3,
                                                 (short)0, acc3, false, false);
  }

  const float bias0 = b1[sub];
  const float bias1 = b1[16 + sub];
  const float bias2 = b1[32 + sub];
  const float bias3 = b1[48 + sub];
#pragma unroll
  for (int j = 0; j < 8; ++j) {
    const size_t base = (size_t)(rowBase + j + 8 * half) * D_OUT + sub;
    float v0 = acc0[j] + bias0;
    float v1 = acc1[j] + bias1;
    float v2 = acc2[j] + bias2;
    float v3 = acc3[j] + bias3;
    h[base]      = v0 > 0.f ? v0 : 0.f;
    h[base + 16] = v1 > 0.f ? v1 : 0.f;
    h[base + 32] = v2 > 0.f ? v2 : 0.f;
    h[base + 48] = v3 > 0.f ? v3 : 0.f;
  }
}

// -------------------------------------------------------------------------
// Kernel 2: source-degree via hardware f32 atomics.
// -------------------------------------------------------------------------
__global__ __launch_bounds__(256)
void fagcn_degree(const int* __restrict__ row, float* __restrict__ deg) {
  int i = blockIdx.x * blockDim.x + threadIdx.x;
  const int stride = gridDim.x * blockDim.x;
  for (int e = i; e < N_EDGES; e += stride) {
    (void)__hip_atomic_fetch_add(&deg[row[e]], 1.0f,
                                 __ATOMIC_RELAXED, __HIP_MEMORY_SCOPE_AGENT);
  }
}

// dnorm[i] = clamp(deg,1)^-0.5, in place
__global__ __launch_bounds__(256)
void fagcn_dnorm(float* __restrict__ deg) {
  int i = blockIdx.x * blockDim.x + threadIdx.x;
  if (i < N_NODES) deg[i] = rsqrtf(fmaxf(deg[i], 1.0f));
}

// -------------------------------------------------------------------------
// Kernel 3: edge phase, one wave32 per edge (grid-stride over edges).
// Lane l owns features {2l, 2l+1}: gathers of h[row]/h[col] are coalesced
// 256B vector loads from the L2-resident h; gate dot reduced with 5 xor-
// shuffles; scatter-add via global_atomic_add_f32.
// -------------------------------------------------------------------------
__global__ __launch_bounds__(256)
void fagcn_edge(const float* __restrict__ h,
                const int*   __restrict__ row,
                const int*   __restrict__ col,
                const float* __restrict__ dnorm,
                const float* __restrict__ gate_w,   // 128 floats, this layer
                const float* __restrict__ gate_b,   // 1 float, this layer
                float*       __restrict__ agg) {
  const int lane   = threadIdx.x & 31;
  const int wave   = (blockIdx.x * blockDim.x + threadIdx.x) >> 5;
  const int nWaves = (gridDim.x * blockDim.x) >> 5;

  const float gb = *gate_b;
  const v2f wa = *(const v2f*)(gate_w + 2 * lane);            // W_gate[:D]
  const v2f wb = *(const v2f*)(gate_w + D_OUT + 2 * lane);    // W_gate[D:]

  for (int e = wave; e < N_EDGES; e += nWaves) {
    const int r = row[e];   // wave-uniform
    const int c = col[e];
    const v2f hr = *(const v2f*)(h + (size_t)r * D_OUT + 2 * lane);
    const v2f hc = *(const v2f*)(h + (size_t)c * D_OUT + 2 * lane);

    float p = hr.x * wa.x + hr.y * wa.y + hc.x * wb.x + hc.y * wb.y;
#pragma unroll
    for (int off = 16; off > 0; off >>= 1)
      p += __shfl_xor(p, off, 32);

    const float g   = tanhf(p + gb);
    const float nrm = g * dnorm[r] * dnorm[c];

    float* dst = agg + (size_t)c * D_OUT + 2 * lane;
    (void)__hip_atomic_fetch_add(dst,     nrm * hr.x,
                                 __ATOMIC_RELAXED, __HIP_MEMORY_SCOPE_AGENT);
    (void)__hip_atomic_fetch_add(dst + 1, nrm * hr.y,
                                 __ATOMIC_RELAXED, __HIP_MEMORY_SCOPE_AGENT);
  }
}

// -------------------------------------------------------------------------
// Kernel 4: h_out = EPS * raw + agg   (streaming)
// -------------------------------------------------------------------------
__global__ __launch_bounds__(256)
void fagcn_combine(const float* __restrict__ raw,
                   const float* __restrict__ agg,
                   float* __restrict__ out) {
  int i = blockIdx.x * blockDim.x + threadIdx.x;
  const int stride = gridDim.x * blockDim.x;
  const int total = N_NODES * D_OUT;
  for (; i < total; i += stride)
    out[i] = EPS_RAW * raw[i] + agg[i];
}

extern "C" void kernel_launch(void* const* d_in, const int* in_sizes, int n_in,
                              void* d_out, int out_size, void* d_ws, size_t ws_size,
                              hipStream_t stream) {
  const float* x   = (const float*)d_in[0];
  const int*   ei  = (const int*)  d_in[1];   // [2, E]: row then col
  const float* W1  = (const float*)d_in[2];   // [64, 256]
  const float* b1  = (const float*)d_in[3];   // [64]
  const float* gw  = (const float*)d_in[4];   // [2, 128]
  const float* gb  = (const float*)d_in[5];   // [2]

  const int* row = ei;
  const int* col = ei + N_EDGES;

  // workspace layout (floats): raw | hbuf | agg | deg   (~77 MB)
  float* raw  = (float*)d_ws;
  float* hbuf = raw  + (size_t)N_NODES * D_OUT;
  float* agg  = hbuf + (size_t)N_NODES * D_OUT;
  float* deg  = agg  + (size_t)N_NODES * D_OUT;

  hipMemsetAsync(deg, 0, (size_t)N_NODES * sizeof(float), stream);

  // raw = relu(x @ W1^T + b1): 6250 row-tile waves -> 782 blocks of 8 waves
  fagcn_gemm_relu<<<782, 256, 0, stream>>>(x, W1, b1, raw);

  fagcn_degree<<<2048, 256, 0, stream>>>(row, deg);
  fagcn_dnorm<<<(N_NODES + 255) / 256, 256, 0, stream>>>(deg);

  const float* hin = raw;
  for (int l = 0; l < N_LAYERS; ++l) {
    hipMemsetAsync(agg, 0, (size_t)N_NODES * D_OUT * sizeof(float), stream);
    fagcn_edge<<<4096, 256, 0, stream>>>(hin, row, col, deg,
                                         gw + (size_t)l * 2 * D_OUT, gb + l, agg);
    float* hout = (l == N_LAYERS - 1) ? (float*)d_out : hbuf;
    fagcn_combine<<<4096, 256, 0, stream>>>(raw, agg, hout);
    hin = hout;
  }
}